// GAT_10213432229996
// MI455X (gfx1250) — compile-verified
//
#include <hip/hip_runtime.h>
#include <cstdint>

typedef __attribute__((ext_vector_type(2))) float v2f;
typedef __attribute__((ext_vector_type(8))) float v8f;

#define GAT_HID 64
#define GAT_H1DIM 512   // HEADS1 * HID

// ---------- helpers ----------
static __device__ __forceinline__ int enc_f32(float f) {
  int i = __float_as_int(f);
  return (i < 0) ? (i ^ 0x7FFFFFFF) : i;   // monotone float -> signed int map
}
static __device__ __forceinline__ float dec_f32(int i) {
  return __int_as_float((i < 0) ? (i ^ 0x7FFFFFFF) : i);
}
static __device__ __forceinline__ void edge_nodes(const int* __restrict__ ei, int e,
                                                  int n_base, int& s, int& d) {
  if (e < n_base) { s = ei[e]; d = ei[n_base + e]; }
  else            { s = e - n_base; d = s; }          // appended self-loops
}

// ---------- fills ----------
__global__ void fill_i32_kernel(int* __restrict__ p, int v, long long n) {
  long long i = (long long)blockIdx.x * blockDim.x + threadIdx.x;
  if (i < n) p[i] = v;
}
__global__ void fill_f32_kernel(float* __restrict__ p, float v, long long n) {
  long long i = (long long)blockIdx.x * blockDim.x + threadIdx.x;
  if (i < n) p[i] = v;
}

// ---------- WMMA f32 GEMM: C[MxN] = A[MxK] * B[KxN], row-major ----------
// One wave computes RT rows x 64 cols (RT/16 x 4 tiles of 16x16).
// N % 64 == 0, K % 4 == 0. RT=32 keeps 8 v8f accumulators (64 VGPRs) --
// fits the register budget without spills while amortizing B fragments
// across two A fragments (2xb64 + 8xb32 loads per 8 WMMAs).
template<int RT>
__global__ void __launch_bounds__(256)
gemm_f32_wmma_kernel(const float* __restrict__ A,
                     const float* __restrict__ B,
                     float* __restrict__ C,
                     int N, int K, int row_base, int n_row_tiles) {
  constexpr int NS = RT / 16;
  const int wave   = (int)((blockIdx.x * blockDim.x + threadIdx.x) >> 5);
  const int lane   = threadIdx.x & 31;
  const int nstrip = N >> 6;
  const int rt = wave / nstrip;
  const int nt = wave % nstrip;
  if (rt >= n_row_tiles) return;            // wave-uniform: EXEC stays all-ones
  const int row0  = row_base + rt * RT;
  const int col0  = nt * 64;
  const int mlane = lane & 15;              // M index (A) / N index (B,C) within tile
  const int khalf = (lane >> 4) << 1;       // lanes 0-15: K={0,1}; 16-31: K={2,3}

  v8f acc[NS][4];
#pragma unroll
  for (int s = 0; s < NS; ++s)
#pragma unroll
    for (int j = 0; j < 4; ++j) acc[s][j] = (v8f){};

  const float* arow = A + (size_t)(row0 + mlane) * K + khalf;
  const size_t astep = (size_t)16 * K;      // next 16-row subtile

  for (int k = 0; k < K; k += 4) {
    v2f a[NS];
#pragma unroll
    for (int s = 0; s < NS; ++s) a[s] = *(const v2f*)(arow + s * astep + k);
    const float* bp = B + (size_t)(k + khalf) * N + col0 + mlane;
    v2f b[4];
#pragma unroll
    for (int j = 0; j < 4; ++j) b[j] = (v2f){ bp[16 * j], bp[16 * j + N] };
#pragma unroll
    for (int s = 0; s < NS; ++s)
#pragma unroll
      for (int j = 0; j < 4; ++j)
        acc[s][j] = __builtin_amdgcn_wmma_f32_16x16x4_f32(
            false, a[s], false, b[j], (short)0, acc[s][j], false, false);
  }

  // D layout: VGPR r: lanes 0-15 -> row r, col=lane; lanes 16-31 -> row 8+r, col=lane-16
  const int rb = (lane >> 4) * 8;
#pragma unroll
  for (int s = 0; s < NS; ++s) {
    float* crow = C + (size_t)(row0 + s * 16) * N + col0 + mlane;
#pragma unroll
    for (int r = 0; r < 8; ++r) {
      float* cp = crow + (size_t)(rb + r) * N;
#pragma unroll
      for (int j = 0; j < 4; ++j) cp[16 * j] = acc[s][j][r];
    }
  }
}

// ---------- attention coefficients: a_src/a_dst [N, heads] ----------
__global__ void att_coef_kernel(const float* __restrict__ H,
                                const float* __restrict__ att_src,
                                const float* __restrict__ att_dst,
                                float* __restrict__ a_src, float* __restrict__ a_dst,
                                int n_nodes, int heads) {
  int i = blockIdx.x * blockDim.x + threadIdx.x;   // i = node*heads + h
  if (i >= n_nodes * heads) return;
  int h = i % heads;
  const float* hp = H + (size_t)i * GAT_HID;       // H is [N, heads*64] contiguous
  const float* as = att_src + h * GAT_HID;
  const float* ad = att_dst + h * GAT_HID;
  float s = 0.f, d = 0.f;
#pragma unroll 8
  for (int f = 0; f < GAT_HID; ++f) { float v = hp[f]; s += v * as[f]; d += v * ad[f]; }
  a_src[i] = s; a_dst[i] = d;
}

// ---------- edge pass 1: segment max of leaky_relu logits ----------
__global__ void edge_max_kernel(const int* __restrict__ ei,
                                const float* __restrict__ a_src,
                                const float* __restrict__ a_dst,
                                int* __restrict__ m_enc,
                                int n_edges_tot, int n_base, int heads) {
  int i = blockIdx.x * blockDim.x + threadIdx.x;
  if (i >= n_edges_tot * heads) return;
  int e = i / heads, h = i - e * heads;
  int s, d; edge_nodes(ei, e, n_base, s, d);
  float l = a_src[s * heads + h] + a_dst[d * heads + h];
  l = (l > 0.f) ? l : 0.2f * l;                    // leaky_relu(0.2)
  atomicMax(&m_enc[d * heads + h], enc_f32(l));
}

// ---------- edge pass 2: p = exp(l - m[dst]); segment sum ----------
__global__ void edge_expsum_kernel(const int* __restrict__ ei,
                                   const float* __restrict__ a_src,
                                   const float* __restrict__ a_dst,
                                   const int* __restrict__ m_enc,
                                   float* __restrict__ p, float* __restrict__ ssum,
                                   int n_edges_tot, int n_base, int heads) {
  int i = blockIdx.x * blockDim.x + threadIdx.x;
  if (i >= n_edges_tot * heads) return;
  int e = i / heads, h = i - e * heads;
  int s, d; edge_nodes(ei, e, n_base, s, d);
  float l = a_src[s * heads + h] + a_dst[d * heads + h];
  l = (l > 0.f) ? l : 0.2f * l;
  float pv = __expf(l - dec_f32(m_enc[d * heads + h]));
  p[i] = pv;
  atomicAdd(&ssum[d * heads + h], pv);
}

// ---------- edge pass 3 (layer 1, 8 heads x 64): scatter-add messages ----------
__global__ void edge_agg1_kernel(const int* __restrict__ ei,
                                 const float* __restrict__ p,
                                 const float* __restrict__ ssum,
                                 const float* __restrict__ H,
                                 float* __restrict__ agg,
                                 int n_edges_tot, int n_base) {
  int wave = (int)((blockIdx.x * blockDim.x + threadIdx.x) >> 5);
  int lane = threadIdx.x & 31;
  if (wave >= n_edges_tot) return;
  int s, d; edge_nodes(ei, wave, n_base, s, d);
  const float* hs = H + (size_t)s * GAT_H1DIM;
  float* ad = agg + (size_t)d * GAT_H1DIM;
#pragma unroll
  for (int it = 0; it < 4; ++it) {
    int idx = it * 128 + lane * 4;                  // 0..508, all 4 elems same head
    int h = idx >> 6;
    float alpha = p[(size_t)wave * 8 + h] / (ssum[d * 8 + h] + 1e-16f);
    float4 v = *(const float4*)(hs + idx);
    atomicAdd(&ad[idx + 0], v.x * alpha);
    atomicAdd(&ad[idx + 1], v.y * alpha);
    atomicAdd(&ad[idx + 2], v.z * alpha);
    atomicAdd(&ad[idx + 3], v.w * alpha);
  }
}

// ---------- edge pass 3 (layer 2, 1 head x 64) ----------
__global__ void edge_agg2_kernel(const int* __restrict__ ei,
                                 const float* __restrict__ p,
                                 const float* __restrict__ ssum,
                                 const float* __restrict__ H,
                                 float* __restrict__ agg,
                                 int n_edges_tot, int n_base) {
  int wave = (int)((blockIdx.x * blockDim.x + threadIdx.x) >> 5);
  int lane = threadIdx.x & 31;
  if (wave >= n_edges_tot) return;
  int s, d; edge_nodes(ei, wave, n_base, s, d);
  float alpha = p[wave] / (ssum[d] + 1e-16f);
  const float* hs = H + (size_t)s * GAT_HID;
  float* ad = agg + (size_t)d * GAT_HID;
  int idx = lane * 2;
  float2 v = *(const float2*)(hs + idx);
  atomicAdd(&ad[idx + 0], v.x * alpha);
  atomicAdd(&ad[idx + 1], v.y * alpha);
}

// ---------- bias + ELU ----------
__global__ void bias_elu_kernel(const float* __restrict__ in, const float* __restrict__ bias,
                                float* __restrict__ out, long long total, int rowlen) {
  long long i = (long long)blockIdx.x * blockDim.x + threadIdx.x;
  if (i >= total) return;
  float v = in[i] + bias[(int)(i % rowlen)];
  out[i] = (v > 0.f) ? v : expm1f(v);
}

// ---------- final FC: logits[n] = H[n,:] . w + b ----------
__global__ void fc_kernel(const float* __restrict__ H, const float* __restrict__ w,
                          const float* __restrict__ b, float* __restrict__ logits, int n) {
  int i = blockIdx.x * blockDim.x + threadIdx.x;
  if (i >= n) return;
  const float* hp = H + (size_t)i * GAT_HID;
  float acc = b[0];
#pragma unroll 8
  for (int f = 0; f < GAT_HID; ++f) acc += hp[f] * w[f];
  logits[i] = acc;
}

// ---------- global softmax over N ----------
__global__ void reduce_max_kernel(const float* __restrict__ x, int n, int* __restrict__ gmax_enc) {
  __shared__ float sm[256];
  float lm = -3.4e38f;
  for (int i = blockIdx.x * blockDim.x + threadIdx.x; i < n; i += gridDim.x * blockDim.x)
    lm = fmaxf(lm, x[i]);
  sm[threadIdx.x] = lm; __syncthreads();
  for (int s = 128; s > 0; s >>= 1) {
    if ((int)threadIdx.x < s) sm[threadIdx.x] = fmaxf(sm[threadIdx.x], sm[threadIdx.x + s]);
    __syncthreads();
  }
  if (threadIdx.x == 0) atomicMax(gmax_enc, enc_f32(sm[0]));
}
__global__ void reduce_sumexp_kernel(const float* __restrict__ x, int n,
                                     const int* __restrict__ gmax_enc, float* __restrict__ gsum) {
  __shared__ float sm[256];
  float gm = dec_f32(*gmax_enc);
  float ls = 0.f;
  for (int i = blockIdx.x * blockDim.x + threadIdx.x; i < n; i += gridDim.x * blockDim.x)
    ls += __expf(x[i] - gm);
  sm[threadIdx.x] = ls; __syncthreads();
  for (int s = 128; s > 0; s >>= 1) {
    if ((int)threadIdx.x < s) sm[threadIdx.x] += sm[threadIdx.x + s];
    __syncthreads();
  }
  if (threadIdx.x == 0) atomicAdd(gsum, sm[0]);
}
__global__ void softmax_out_kernel(const float* __restrict__ x, int n,
                                   const int* __restrict__ gmax_enc,
                                   const float* __restrict__ gsum, float* __restrict__ out) {
  int i = blockIdx.x * blockDim.x + threadIdx.x;
  if (i >= n) return;
  out[i] = __expf(x[i] - dec_f32(*gmax_enc)) / (*gsum);
}

// ---------- host launch ----------
static inline int cdiv_ll(long long a, long long b) { return (int)((a + b - 1) / b); }

static void launch_gemm(const float* A, const float* Bm, float* C,
                        int M, int N, int K, hipStream_t stream) {
  const int B = 256;
  const int nstrip = N / 64;
  const int main_tiles = M / 32;                     // 32-row register-blocked waves
  if (main_tiles > 0) {
    long long waves = (long long)main_tiles * nstrip;
    gemm_f32_wmma_kernel<32><<<cdiv_ll(waves * 32, B), B, 0, stream>>>(
        A, Bm, C, N, K, 0, main_tiles);
  }
  const int rem = M - main_tiles * 32;               // multiple of 16 (tail rows)
  if (rem > 0) {
    const int tail_tiles = rem / 16;
    long long waves = (long long)tail_tiles * nstrip;
    gemm_f32_wmma_kernel<16><<<cdiv_ll(waves * 32, B), B, 0, stream>>>(
        A, Bm, C, N, K, main_tiles * 32, tail_tiles);
  }
}

extern "C" void kernel_launch(void* const* d_in, const int* in_sizes, int n_in,
                              void* d_out, int out_size, void* d_ws, size_t ws_size,
                              hipStream_t stream) {
  const float* x        = (const float*)d_in[0];
  const float* W1       = (const float*)d_in[1];
  const float* att_src1 = (const float*)d_in[2];
  const float* att_dst1 = (const float*)d_in[3];
  const float* b1       = (const float*)d_in[4];
  const float* W2       = (const float*)d_in[5];
  const float* att_src2 = (const float*)d_in[6];
  const float* att_dst2 = (const float*)d_in[7];
  const float* b2       = (const float*)d_in[8];
  const float* fc_w     = (const float*)d_in[9];
  const float* fc_b     = (const float*)d_in[10];
  const int*   ei       = (const int*)d_in[11];

  const int N  = in_sizes[0] / 128;        // 50000 nodes
  const int E0 = in_sizes[11] / 2;         // 800000 base edges
  const int ET = E0 + N;                   // + self loops
  const int H1 = 8, H2 = 1;

  // workspace carving (256B aligned)
  size_t off = 0;
  auto carve = [&](size_t bytes) { size_t o = off; off += (bytes + 255) & ~(size_t)255; return o; };
  char* ws = (char*)d_ws;
  float* h1    = (float*)(ws + carve((size_t)N * GAT_H1DIM * 4));  // X@W1; later reused as elu output
  float* agg1  = (float*)(ws + carve((size_t)N * GAT_H1DIM * 4));
  float* asrc1 = (float*)(ws + carve((size_t)N * H1 * 4));
  float* adst1 = (float*)(ws + carve((size_t)N * H1 * 4));
  int*   m1    = (int*)  (ws + carve((size_t)N * H1 * 4));
  float* s1    = (float*)(ws + carve((size_t)N * H1 * 4));
  float* p1    = (float*)(ws + carve((size_t)ET * H1 * 4));
  float* h2    = (float*)(ws + carve((size_t)N * GAT_HID * 4));
  float* agg2  = (float*)(ws + carve((size_t)N * GAT_HID * 4));
  float* asrc2 = (float*)(ws + carve((size_t)N * H2 * 4));
  float* adst2 = (float*)(ws + carve((size_t)N * H2 * 4));
  int*   m2    = (int*)  (ws + carve((size_t)N * H2 * 4));
  float* s2    = (float*)(ws + carve((size_t)N * H2 * 4));
  float* p2    = (float*)(ws + carve((size_t)ET * H2 * 4));
  float* logit = (float*)(ws + carve((size_t)N * 4));
  int*   gmax  = (int*)  (ws + carve(16));
  float* gsum  = (float*)(gmax + 1);

  const int B = 256;
  const int INT_MIN_V = (int)0x80000000;

  // init
  fill_i32_kernel<<<cdiv_ll((long long)N*H1,B),B,0,stream>>>(m1, INT_MIN_V, (long long)N*H1);
  fill_f32_kernel<<<cdiv_ll((long long)N*H1,B),B,0,stream>>>(s1, 0.f, (long long)N*H1);
  fill_f32_kernel<<<cdiv_ll((long long)N*GAT_H1DIM,B),B,0,stream>>>(agg1, 0.f, (long long)N*GAT_H1DIM);
  fill_i32_kernel<<<cdiv_ll((long long)N*H2,B),B,0,stream>>>(m2, INT_MIN_V, (long long)N*H2);
  fill_f32_kernel<<<cdiv_ll((long long)N*H2,B),B,0,stream>>>(s2, 0.f, (long long)N*H2);
  fill_f32_kernel<<<cdiv_ll((long long)N*GAT_HID,B),B,0,stream>>>(agg2, 0.f, (long long)N*GAT_HID);
  fill_i32_kernel<<<1,32,0,stream>>>(gmax, INT_MIN_V, 1);
  fill_f32_kernel<<<1,32,0,stream>>>(gsum, 0.f, 1);

  // layer 1: GEMM (WMMA f32), attention, segment softmax, aggregation
  launch_gemm(x, W1, h1, N, GAT_H1DIM, 128, stream);
  att_coef_kernel<<<cdiv_ll((long long)N*H1,B),B,0,stream>>>(h1, att_src1, att_dst1, asrc1, adst1, N, H1);
  edge_max_kernel<<<cdiv_ll((long long)ET*H1,B),B,0,stream>>>(ei, asrc1, adst1, m1, ET, E0, H1);
  edge_expsum_kernel<<<cdiv_ll((long long)ET*H1,B),B,0,stream>>>(ei, asrc1, adst1, m1, p1, s1, ET, E0, H1);
  edge_agg1_kernel<<<cdiv_ll((long long)ET*32,B),B,0,stream>>>(ei, p1, s1, h1, agg1, ET, E0);
  bias_elu_kernel<<<cdiv_ll((long long)N*GAT_H1DIM,B),B,0,stream>>>(agg1, b1, h1, (long long)N*GAT_H1DIM, GAT_H1DIM);

  // layer 2
  launch_gemm(h1, W2, h2, N, GAT_HID, GAT_H1DIM, stream);
  att_coef_kernel<<<cdiv_ll((long long)N*H2,B),B,0,stream>>>(h2, att_src2, att_dst2, asrc2, adst2, N, H2);
  edge_max_kernel<<<cdiv_ll((long long)ET*H2,B),B,0,stream>>>(ei, asrc2, adst2, m2, ET, E0, H2);
  edge_expsum_kernel<<<cdiv_ll((long long)ET*H2,B),B,0,stream>>>(ei, asrc2, adst2, m2, p2, s2, ET, E0, H2);
  edge_agg2_kernel<<<cdiv_ll((long long)ET*32,B),B,0,stream>>>(ei, p2, s2, h2, agg2, ET, E0);
  bias_elu_kernel<<<cdiv_ll((long long)N*GAT_HID,B),B,0,stream>>>(agg2, b2, agg2, (long long)N*GAT_HID, GAT_HID);

  // FC + global softmax over nodes
  fc_kernel<<<cdiv_ll(N,B),B,0,stream>>>(agg2, fc_w, fc_b, logit, N);
  reduce_max_kernel<<<256,B,0,stream>>>(logit, N, gmax);
  reduce_sumexp_kernel<<<256,B,0,stream>>>(logit, N, gmax, gsum);
  softmax_out_kernel<<<cdiv_ll(N,B),B,0,stream>>>(logit, N, gmax, gsum, (float*)d_out);
}